// GNN_13786845021020
// MI455X (gfx1250) — compile-verified
//
#include <hip/hip_runtime.h>

typedef __attribute__((ext_vector_type(2))) float v2f;
typedef __attribute__((ext_vector_type(8))) float v8f;

#define D 128            // feature width (both layers)
#define NT 8             // 8 N-tiles of 16 -> 128 output cols
#define BLOCK_M 128      // rows per workgroup (8 waves x 16 rows)
#define PITCH 132        // LDS pitch (dwords): 132 % 64 == 4 -> conflict-free b64 frag loads
#define USE_ASYNC_LDS 1  // stage A tile via global_load_async_to_lds_b128

// ---------------------------------------------------------------- utilities
__global__ void zero_kernel(float4* __restrict__ p, int n4) {
    int i = blockIdx.x * blockDim.x + threadIdx.x;
    if (i < n4) p[i] = make_float4(0.f, 0.f, 0.f, 0.f);
}

__global__ void degree_kernel(const int* __restrict__ dst, float* __restrict__ deg, int n_edges) {
    int e = blockIdx.x * blockDim.x + threadIdx.x;
    if (e < n_edges) atomicAdd(&deg[dst[e]], 1.0f);
}

__global__ void invdeg_kernel(float* __restrict__ deg, int n) {
    int i = blockIdx.x * blockDim.x + threadIdx.x;
    if (i < n) deg[i] = 1.0f / fmaxf(deg[i], 1.0f);
}

// One wave32 per edge; lane handles 4 consecutive floats (coalesced 512B gather,
// 4x global_atomic_add_f32 scatter; agg fits in 192MB L2 so atomics resolve on-chip).
__global__ void scatter_kernel(const int* __restrict__ src, const int* __restrict__ dst,
                               const float* __restrict__ feat, float* __restrict__ agg,
                               int n_edges) {
    int gtid = blockIdx.x * blockDim.x + threadIdx.x;
    int e    = gtid >> 5;
    int lane = gtid & 31;
    if (e >= n_edges) return;
    int s = src[e];
    int d = dst[e];
    const float4 v = *(const float4*)(feat + (size_t)s * D + lane * 4);
    float* b = agg + (size_t)d * D + lane * 4;
    atomicAdd(b + 0, v.x);
    atomicAdd(b + 1, v.y);
    atomicAdd(b + 2, v.z);
    atomicAdd(b + 3, v.w);
}

// ------------------------------------------------------- fused SAGE GEMM
// out[i,:] = act( Aself[i,:] @ Wself + (invdeg[i] * Aagg[i,:]) @ Wneigh + bias )
// One K=256 GEMM (two 128-wide halves) with fp32 WMMA 16x16x4, register-pipelined.
__global__ void sage_gemm_kernel(const float* __restrict__ Aself,
                                 const float* __restrict__ Aagg,
                                 const float* __restrict__ invdeg,
                                 const float* __restrict__ Wself,
                                 const float* __restrict__ Wneigh,
                                 const float* __restrict__ bias,
                                 float* __restrict__ out,
                                 int n_rows, int do_relu) {
    extern __shared__ float smem[];
    float* As = smem;                   // [BLOCK_M][PITCH], current K-half of A
    float* Bs = smem + BLOCK_M * PITCH; // [128 n][PITCH], current K-half of W, transposed

    const int tid     = threadIdx.x;
    const int lane    = tid & 31;
    const int wave    = tid >> 5;
    const int l15     = lane & 15;
    const int khalf   = lane >> 4;      // 0: K pair {0,1}; 1: K pair {2,3}
    const int kpair   = khalf << 1;
    const int rowbase = blockIdx.x * BLOCK_M;

    // Per-lane A-fragment row (constant across k loop) and its neighbor-scale.
    const int arow_g  = rowbase + wave * 16 + l15;
    const int arow_c  = (arow_g < n_rows) ? arow_g : (n_rows - 1);
    const float nscale = invdeg[arow_c];

    v8f acc[NT];
#pragma unroll
    for (int t = 0; t < NT; ++t) acc[t] = (v8f){0.f, 0.f, 0.f, 0.f, 0.f, 0.f, 0.f, 0.f};

    const float* aptr = &As[(wave * 16 + l15) * PITCH + kpair];

    for (int half = 0; half < 2; ++half) {
        __syncthreads();  // previous half's As/Bs no longer being read

        // ---- Stage A half-block [BLOCK_M x 128] into LDS ----
        const float* Ain = half ? Aagg : Aself;
#if USE_ASYNC_LDS
        for (int idx = tid; idx < BLOCK_M * (D / 4); idx += blockDim.x) {
            int r    = idx >> 5;            // D/4 == 32
            int c4   = (idx & 31) << 2;
            int grow = rowbase + r;
            if (grow >= n_rows) grow = n_rows - 1;   // clamp: dup rows never stored
            const float* gptr = Ain + (size_t)grow * D + c4;
            unsigned ldsoff = (unsigned)(uintptr_t)(&As[r * PITCH + c4]);
            asm volatile("global_load_async_to_lds_b128 %0, %1, off"
                         :: "v"(ldsoff), "v"(gptr) : "memory");
        }
#else
        for (int idx = tid; idx < BLOCK_M * (D / 4); idx += blockDim.x) {
            int r    = idx >> 5;
            int c4   = (idx & 31) << 2;
            int grow = rowbase + r;
            if (grow >= n_rows) grow = n_rows - 1;
            *(float4*)&As[r * PITCH + c4] = *(const float4*)(Ain + (size_t)grow * D + c4);
        }
#endif
        // ---- Stage W half transposed: Bs[n][k] (overlaps with async A DMA) ----
        const float* W = half ? Wneigh : Wself;
        for (int idx = tid; idx < D * D; idx += blockDim.x) {
            int k = idx >> 7;       // 0..127
            int n = idx & (D - 1);  // 0..127
            Bs[n * PITCH + k] = W[k * D + n];
        }
#if USE_ASYNC_LDS
        asm volatile("s_wait_asynccnt 0x0" ::: "memory");
#endif
        __syncthreads();

        // ---- Compute: two-stage register pipeline over the 128-deep half ----
        const float s = half ? nscale : 1.0f;
        v2f a0, a1, b0[NT], b1[NT];

        a0 = *(const v2f*)(aptr + 0);
#pragma unroll
        for (int t = 0; t < NT; ++t)
            b0[t] = *(const v2f*)&Bs[(t * 16 + l15) * PITCH + kpair];

        for (int kk = 0; kk < D; kk += 8) {
            // issue loads for kk+4 into shadow set
            a1 = *(const v2f*)(aptr + kk + 4);
#pragma unroll
            for (int t = 0; t < NT; ++t)
                b1[t] = *(const v2f*)&Bs[(t * 16 + l15) * PITCH + kk + 4 + kpair];

            // 8 back-to-back WMMAs on current set (kk)
            v2f as; as[0] = a0[0] * s; as[1] = a0[1] * s;
#pragma unroll
            for (int t = 0; t < NT; ++t)
                acc[t] = __builtin_amdgcn_wmma_f32_16x16x4_f32(
                    false, as, false, b0[t], (short)0, acc[t], false, false);

            // issue loads for kk+8 into primary set
            if (kk + 8 < D) {
                a0 = *(const v2f*)(aptr + kk + 8);
#pragma unroll
                for (int t = 0; t < NT; ++t)
                    b0[t] = *(const v2f*)&Bs[(t * 16 + l15) * PITCH + kk + 8 + kpair];
            }

            // 8 WMMAs on shadow set (kk+4)
            v2f as1; as1[0] = a1[0] * s; as1[1] = a1[1] * s;
#pragma unroll
            for (int t = 0; t < NT; ++t)
                acc[t] = __builtin_amdgcn_wmma_f32_16x16x4_f32(
                    false, as1, false, b1[t], (short)0, acc[t], false, false);
        }
    }

    // Epilogue: C/D layout -> lane n = t*16+l15, rows mbase..mbase+7 in 8 VGPRs
    const int mbase = rowbase + wave * 16 + (khalf << 3);
#pragma unroll
    for (int t = 0; t < NT; ++t) {
        int col  = t * 16 + l15;
        float bv = bias[col];
#pragma unroll
        for (int r = 0; r < 8; ++r) {
            int grow = mbase + r;
            if (grow < n_rows) {
                float v = acc[t][r] + bv;
                if (do_relu) v = fmaxf(v, 0.0f);
                out[(size_t)grow * D + col] = v;
            }
        }
    }
}

// ------------------------------------------------------ classifier (128 -> 2)
__global__ void classifier_kernel(const float* __restrict__ h, const float* __restrict__ Wc,
                                  const float* __restrict__ bc, float* __restrict__ out, int n) {
    int gtid = blockIdx.x * blockDim.x + threadIdx.x;
    int node = gtid >> 5;
    int lane = gtid & 31;
    if (node >= n) return;
    const float4 v = *(const float4*)(h + (size_t)node * D + lane * 4);
    int kb = lane * 4;
    float c0 = 0.f, c1 = 0.f;
    const float hv[4] = {v.x, v.y, v.z, v.w};
#pragma unroll
    for (int j = 0; j < 4; ++j) {
        c0 += hv[j] * Wc[(kb + j) * 2 + 0];
        c1 += hv[j] * Wc[(kb + j) * 2 + 1];
    }
#pragma unroll
    for (int off = 16; off > 0; off >>= 1) {
        c0 += __shfl_down(c0, off, 32);
        c1 += __shfl_down(c1, off, 32);
    }
    if (lane == 0) {
        out[node * 2 + 0] = c0 + bc[0];
        out[node * 2 + 1] = c1 + bc[1];
    }
}

// ---------------------------------------------------------------- launcher
extern "C" void kernel_launch(void* const* d_in, const int* in_sizes, int n_in,
                              void* d_out, int out_size, void* d_ws, size_t ws_size,
                              hipStream_t stream) {
    const float* feat = (const float*)d_in[0];
    const int*   src  = (const int*)d_in[1];
    const int*   dst  = (const int*)d_in[2];
    const float* W1s  = (const float*)d_in[3];
    const float* W1n  = (const float*)d_in[4];
    const float* b1   = (const float*)d_in[5];
    const float* W2s  = (const float*)d_in[6];
    const float* W2n  = (const float*)d_in[7];
    const float* b2   = (const float*)d_in[8];
    const float* Wc   = (const float*)d_in[9];
    const float* bc   = (const float*)d_in[10];
    float* out = (float*)d_out;

    const int n_nodes = in_sizes[0] / D;  // 100000
    const int n_edges = in_sizes[1];      // 1600000

    // Workspace carve-up (floats): deg | agg | h1 | h2
    float* ws  = (float*)d_ws;
    size_t pad = ((size_t)n_nodes + 1023) & ~(size_t)1023;
    float* deg = ws;
    float* agg = ws + pad;
    float* h1  = agg + (size_t)n_nodes * D;
    float* h2  = h1 + (size_t)n_nodes * D;

    const int  T = 256;
    const int  feat4   = n_nodes * (D / 4);
    const int  deg4    = (n_nodes + 3) / 4;
    const int  gemmBlk = (n_nodes + BLOCK_M - 1) / BLOCK_M; // 782
    const size_t smem  = (size_t)(2 * BLOCK_M * PITCH) * sizeof(float); // ~135KB -> 2 WG/WGP

    // Degrees -> inv_deg (in place)
    zero_kernel<<<(deg4 + T - 1) / T, T, 0, stream>>>((float4*)deg, deg4);
    degree_kernel<<<(n_edges + T - 1) / T, T, 0, stream>>>(dst, deg, n_edges);
    invdeg_kernel<<<(n_nodes + T - 1) / T, T, 0, stream>>>(deg, n_nodes);

    // Layer 1: aggregate features, fused GEMM + bias + ReLU
    zero_kernel<<<(feat4 + T - 1) / T, T, 0, stream>>>((float4*)agg, feat4);
    scatter_kernel<<<(n_edges * 32 + T - 1) / T, T, 0, stream>>>(src, dst, feat, agg, n_edges);
    sage_gemm_kernel<<<gemmBlk, T, smem, stream>>>(feat, agg, deg, W1s, W1n, b1, h1, n_nodes, 1);

    // Layer 2: aggregate h1, fused GEMM + bias (no ReLU)
    zero_kernel<<<(feat4 + T - 1) / T, T, 0, stream>>>((float4*)agg, feat4);
    scatter_kernel<<<(n_edges * 32 + T - 1) / T, T, 0, stream>>>(src, dst, h1, agg, n_edges);
    sage_gemm_kernel<<<gemmBlk, T, smem, stream>>>(h1, agg, deg, W2s, W2n, b2, h2, n_nodes, 0);

    // Classifier
    classifier_kernel<<<(n_nodes * 32 + T - 1) / T, T, 0, stream>>>(h2, Wc, bc, out, n_nodes);
}